// GCN_46308337386171
// MI455X (gfx1250) — compile-verified
//
#include <hip/hip_runtime.h>

// ---------------------------------------------------------------------------
// 2-layer GCN for MI455X (gfx1250, wave32).
// GEMMs use V_WMMA_F32_16X16X4_F32 (fp32-exact). Aggregation is the roofline
// bottleneck (~150M f32 atomics, h-tables L2-resident) and uses coalesced
// float4 gathers + hardware global_atomic_add_f32 scatters with all lanes hot.
// ---------------------------------------------------------------------------

typedef __attribute__((ext_vector_type(2))) float v2f;
typedef __attribute__((ext_vector_type(8))) float v8f;

#define K_DIM 128         // in_c == hid_c == 128 for both layers
#define LDS_STRIDE 132    // pad 128 -> 132 floats to avoid LDS bank conflicts

// ---------------- degree / norm precompute ----------------

__global__ void k_deg_init(int* __restrict__ deg, int n) {
    int i = blockIdx.x * blockDim.x + threadIdx.x;
    if (i < n) deg[i] = 1;  // self-loop
}

__global__ void k_deg_count(const long long* __restrict__ dst, int* __restrict__ deg, int E) {
    int e = blockIdx.x * blockDim.x + threadIdx.x;
    if (e < E) atomicAdd(&deg[(int)dst[e]], 1);
}

// in-place: reinterpret int degree -> float d^{-1/2}
__global__ void k_dinv(float* __restrict__ dinv, int n) {
    int i = blockIdx.x * blockDim.x + threadIdx.x;
    if (i < n) {
        int d = ((const int*)dinv)[i];
        dinv[i] = rsqrtf((float)d);  // d >= 1 always (self-loop)
    }
}

__global__ void k_norm(const long long* __restrict__ src, const long long* __restrict__ dst,
                       const float* __restrict__ dinv, float* __restrict__ norm, int E) {
    int e = blockIdx.x * blockDim.x + threadIdx.x;
    if (e < E) norm[e] = dinv[(int)src[e]] * dinv[(int)dst[e]];
}

// ---------------- WMMA fp32 GEMM: out[n_rows x COUT] = act(A[n_rows x 128]) @ W[128 x COUT] ----

template <int COUT, bool RELU>
__global__ __launch_bounds__(COUT * 2) void k_gemm(const float* __restrict__ A,
                                                   const float* __restrict__ W,
                                                   float* __restrict__ out,
                                                   int n_rows) {
    __shared__ float as[16 * LDS_STRIDE];
    const int row0 = blockIdx.x * 16;
    const int tid = threadIdx.x;
    const int nthreads = COUT * 2;  // COUT/16 waves of 32

    // Stage 16 x 128 A-tile into LDS (float4 loads), fusing ReLU for layer 2.
    for (int i = tid; i < 16 * (K_DIM / 4); i += nthreads) {
        int r = i >> 5;        // K_DIM/4 == 32
        int c4 = i & 31;
        float4 v = make_float4(0.f, 0.f, 0.f, 0.f);
        if (row0 + r < n_rows)
            v = ((const float4*)(A + (size_t)(row0 + r) * K_DIM))[c4];
        if (RELU) {
            v.x = fmaxf(v.x, 0.f); v.y = fmaxf(v.y, 0.f);
            v.z = fmaxf(v.z, 0.f); v.w = fmaxf(v.w, 0.f);
        }
        float* p = &as[r * LDS_STRIDE + c4 * 4];
        p[0] = v.x; p[1] = v.y; p[2] = v.z; p[3] = v.w;
    }
    __syncthreads();

    const int wave = tid >> 5;        // one 16x16 output tile per wave
    const int lane = tid & 31;
    const int colblk = wave * 16;
    const int m  = lane & 15;         // row (A) / col (B,C)
    const int kh = lane >> 4;         // K-half selector

    v8f acc = {};
    const float* wcol = W + colblk + m;  // W is [128 x COUT] row-major

    // A 16x4 f32 layout: lane = M + 16*(K/2), vgpr = K%2
    // B  4x16 f32 layout: vgpr v -> K = 2v + (lane/16), N = lane%16
#pragma unroll 4
    for (int k0 = 0; k0 < K_DIM; k0 += 4) {
        const float* ap = &as[m * LDS_STRIDE + k0 + 2 * kh];
        v2f a;
        a.x = ap[0];
        a.y = ap[1];
        v2f b;
        b.x = wcol[(size_t)(k0 + kh) * COUT];
        b.y = wcol[(size_t)(k0 + 2 + kh) * COUT];
        acc = __builtin_amdgcn_wmma_f32_16x16x4_f32(false, a, false, b,
                                                    (short)0, acc, false, false);
    }

    // C/D layout: vgpr g -> row = g + 8*(lane/16), col = lane%16
    if (row0 + 16 <= n_rows) {
        // full tile (always taken for N % 16 == 0): unguarded, clause-able stores
        float* outp = out + (size_t)(row0 + 8 * kh) * COUT + colblk + m;
#pragma unroll
        for (int g = 0; g < 8; ++g)
            outp[(size_t)g * COUT] = acc[g];
    } else {
#pragma unroll
        for (int g = 0; g < 8; ++g) {
            int r = row0 + g + 8 * kh;
            if (r < n_rows)
                out[(size_t)r * COUT + colblk + m] = acc[g];
        }
    }
}

// ---------------- aggregation ----------------

// out[v][c] = bias[c] + dinv[v]^2 * h[v][c]   (self-loop term + bias), float4-wide
template <int C>
__global__ void k_agg_init(const float* __restrict__ h, const float* __restrict__ dinv,
                           const float* __restrict__ bias, float* __restrict__ out, int n) {
    constexpr int C4 = C / 4;
    int i = blockIdx.x * blockDim.x + threadIdx.x;  // one float4 per thread
    if (i < n * C4) {
        int v = i / C4;
        int c4 = i % C4;
        float di = dinv[v];
        float s = di * di;
        float4 hv = ((const float4*)h)[i];
        float4 bv = ((const float4*)bias)[c4];
        float4 o;
        o.x = bv.x + s * hv.x; o.y = bv.y + s * hv.y;
        o.z = bv.z + s * hv.z; o.w = bv.w + s * hv.w;
        ((float4*)out)[i] = o;
    }
}

// C/4 lanes per edge (32 for C=128, 16 for C=64 -> 2 edges/wave, all lanes hot):
// coalesced float4 gather of h[src], 4 HW float atomics into out[dst].
template <int C>
__global__ void k_agg_edges(const float* __restrict__ h, const float* __restrict__ norm,
                            const long long* __restrict__ src, const long long* __restrict__ dst,
                            float* __restrict__ out, int E) {
    constexpr int LPE = C / 4;  // lanes per edge (power of 2)
    long long tid = (long long)blockIdx.x * blockDim.x + threadIdx.x;
    int e   = (int)(tid / LPE);
    int sub = (int)(tid % LPE);
    if (e >= E) return;
    int s = (int)src[e];
    int d = (int)dst[e];
    float w = norm[e];
    float4 v = ((const float4*)(h + (size_t)s * C))[sub];
    float* op = out + (size_t)d * C + sub * 4;
    unsafeAtomicAdd(op + 0, v.x * w);
    unsafeAtomicAdd(op + 1, v.y * w);
    unsafeAtomicAdd(op + 2, v.z * w);
    unsafeAtomicAdd(op + 3, v.w * w);
}

// ---------------- launch ----------------

extern "C" void kernel_launch(void* const* d_in, const int* in_sizes, int n_in,
                              void* d_out, int out_size, void* d_ws, size_t ws_size,
                              hipStream_t stream) {
    const float*     x  = (const float*)d_in[0];
    const long long* ei = (const long long*)d_in[1];  // int64 edge_index [2, E]
    const float*     W1 = (const float*)d_in[2];
    const float*     b1 = (const float*)d_in[3];
    const float*     W2 = (const float*)d_in[4];
    const float*     b2 = (const float*)d_in[5];
    float* out = (float*)d_out;

    const int N = in_sizes[0] / 128;
    const int E = in_sizes[1] / 2;
    const long long* e_src = ei;
    const long long* e_dst = ei + E;

    // workspace carve-up (256B aligned)
    char* ws = (char*)d_ws;
    size_t off = 0;
    auto carve = [&](size_t bytes) -> void* {
        void* p = ws + off;
        off = (off + bytes + 255) & ~(size_t)255;
        return p;
    };
    float* dinv  = (float*)carve((size_t)N * 4);        // int degree, then float d^-1/2
    float* norm  = (float*)carve((size_t)E * 4);
    float* h1    = (float*)carve((size_t)N * 128 * 4);  // x @ W1
    float* h1a   = (float*)carve((size_t)N * 128 * 4);  // aggregated + bias (pre-relu)
    float* h2    = (float*)carve((size_t)N * 64 * 4);   // relu(h1a) @ W2
    (void)ws_size; (void)n_in; (void)out_size;

    const int T = 256;
    // 1. degrees (exact int atomics), then d^-1/2, then per-edge norm
    k_deg_init<<<(N + T - 1) / T, T, 0, stream>>>((int*)dinv, N);
    k_deg_count<<<(E + T - 1) / T, T, 0, stream>>>(e_dst, (int*)dinv, E);
    k_dinv<<<(N + T - 1) / T, T, 0, stream>>>(dinv, N);
    k_norm<<<(E + T - 1) / T, T, 0, stream>>>(e_src, e_dst, dinv, norm, E);

    const int rowBlocks = (N + 15) / 16;

    // 2. layer 1: h1 = x @ W1 ; h1a = b1 + dinv^2*h1 + scatter(norm * h1[src])
    k_gemm<128, false><<<rowBlocks, 256, 0, stream>>>(x, W1, h1, N);
    k_agg_init<128><<<((size_t)N * 32 + T - 1) / T, T, 0, stream>>>(h1, dinv, b1, h1a, N);
    k_agg_edges<128><<<((size_t)E * 32 + T - 1) / T, T, 0, stream>>>(h1, norm, e_src, e_dst, h1a, E);

    // 3. layer 2: h2 = relu(h1a) @ W2 ; out = b2 + dinv^2*h2 + scatter(norm * h2[src])
    k_gemm<64, true><<<rowBlocks, 128, 0, stream>>>(h1a, W2, h2, N);
    k_agg_init<64><<<((size_t)N * 16 + T - 1) / T, T, 0, stream>>>(h2, dinv, b2, out, N);
    k_agg_edges<64><<<((size_t)E * 16 + T - 1) / T, T, 0, stream>>>(h2, norm, e_src, e_dst, out, E);
}